// BidirectionalTrustModel_10041633538061
// MI455X (gfx1250) — compile-verified
//
#include <hip/hip_runtime.h>
#include <hip/hip_bf16.h>

typedef __attribute__((ext_vector_type(2))) float v2f;
typedef __attribute__((ext_vector_type(8))) float v8f;

#define BINS_ 10
#define T_ 64
#define NTOT_ 65536
#define BLK_ 256

// One thread per n. Streaming scan over T (bandwidth-bound: ~64.5 MB @ 23.3 TB/s
// => ~2.8 us floor), then WMMA f32 16x16x4 epilogue computing the bin reductions
//   s_i = sum_k Dtab[b_n][k] * m_i[k]   and   c_i = sum_k m_i[k]
// for 16 columns per tile (rows 0..9 = d-table rows, row 10 = ones => counts).
__global__ __launch_bounds__(BLK_) void trust_kernel(
    const int*   __restrict__ perf,      // [T,N,2] int32
    const float* __restrict__ obs_sens,  // [T,N]
    const float* __restrict__ obs_proc,  // [T,N]
    const float* __restrict__ pred_sens, // [N,1]
    const float* __restrict__ pred_proc, // [N,1]
    const float* __restrict__ betas,     // [2]
    const float* __restrict__ zetas,     // [2]
    float* __restrict__ out)             // [N,1]
{
    __shared__ float Aext[2][16][12];   // extended d-table per channel
    __shared__ float dump[8][256];      // per-wave WMMA result scratch

    const int tid  = threadIdx.x;
    const int lane = tid & 31;
    const int wave = tid >> 5;
    const int h    = lane >> 4;   // wave half
    const int c    = lane & 15;   // column / M-row index within tile
    const int n    = blockIdx.x * BLK_ + tid;

    // ---- Phase 0: build extended d-tables in LDS (2 x 16 x 12) ----
    for (int idx = tid; idx < 2 * 16 * 12; idx += BLK_) {
        const int ch  = idx / (16 * 12);
        const int rem = idx % (16 * 12);
        const int M   = rem / 12;
        const int k   = rem % 12;
        float v = 0.0f;
        if (M < 10 && k < 10) {
            const float req  = ((float)M + 0.5f) / 10.0f;       // grid value (division, as in dataset)
            const float step = ((float)k + 0.5f) * 0.1f;        // bin center (multiply, as in reference)
            const float beta = betas[ch];
            const float z    = zetas[ch];
            v = powf(1.0f + expf(beta * (req - step)), -(z * z));
        } else if (M == 10 && k < 10) {
            v = 1.0f;                                           // count row
        }
        Aext[ch][M][k] = v;
    }
    __syncthreads();

    // Preload A fragments (A is 16x4 per chunk: lane holds M=c; VGPR j holds K=4q+2h+j)
    float afrag[2][3][2];
#pragma unroll
    for (int ch = 0; ch < 2; ++ch)
#pragma unroll
        for (int q = 0; q < 3; ++q)
#pragma unroll
            for (int j = 0; j < 2; ++j)
                afrag[ch][q][j] = Aext[ch][c][4 * q + 2 * h + j];

    // ---- Phase 1: sequential interval scan over T (bit-exact vs reference) ----
    const float inv = 0.1f;
    float low[2]  = {0.0f, 0.0f};
    float high[2] = {1.0f, 1.0f};

#pragma unroll 4
    for (int t = 0; t < T_; ++t) {
        const int base = t * NTOT_ + n;
        if (t + 8 < T_) {
            const int pre = (t + 8) * NTOT_ + n;
            __builtin_prefetch(&obs_sens[pre], 0, 3);
            __builtin_prefetch(&obs_proc[pre], 0, 3);
            __builtin_prefetch(&((const int2*)perf)[pre], 0, 3);
        }
        const int2 p   = ((const int2*)perf)[base];  // (fail bit, success bit)
        const bool succ = (p.x == 0) && (p.y != 0);
        const bool fail = (p.x != 0) && (p.y == 0);
        const float caps[2] = { obs_sens[base], obs_proc[base] };
#pragma unroll
        for (int i = 0; i < 2; ++i) {
            const float lo = low[i], hi = high[i], cp = caps[i];
            const float s_hi = (cp > hi) ? cp : hi;
            const float s_lo = (cp <= hi && cp > lo) ? cp : lo;
            const float f_lo = (cp < lo) ? cp : lo;
            const float f_hi = (cp >= lo && cp < hi) ? cp : hi;
            const float lo2 = succ ? s_lo : (fail ? f_lo : lo);
            const float hi2 = succ ? s_hi : (fail ? f_hi : hi);
            const bool  eq  = (lo2 == hi2);
            high[i] = (eq && hi2 == 0.0f) ? inv : hi2;
            low[i]  = (eq && hi2 != 0.0f) ? (hi2 - inv) : lo2;
        }
    }

    // ---- Phase 2: WMMA epilogue ----
    const int bidx[2] = { (int)(pred_sens[n] * 10.0f), (int)(pred_proc[n] * 10.0f) };

    float ratio[2] = {0.0f, 0.0f};
#pragma unroll
    for (int tile = 0; tile < 2; ++tile) {
#pragma unroll
        for (int ch = 0; ch < 2; ++ch) {
            const int src   = tile * 16 + c;                 // lane owning column c of this tile
            const float locol = __shfl(low[ch],  src, 32);
            const float hicol = __shfl(high[ch], src, 32);
            const int   bcol  = __shfl(bidx[ch], src, 32);

            v8f acc = {};
#pragma unroll
            for (int q = 0; q < 3; ++q) {
                const int   k0    = 4 * q + 2 * h;           // K indices this lane supplies
                const float step0 = ((float)k0 + 0.5f) * 0.1f;
                const float step1 = ((float)(k0 + 1) + 0.5f) * 0.1f;
                const float b0 = (k0     < 10 && step0 >= locol && step0 <= hicol) ? 1.0f : 0.0f;
                const float b1 = (k0 + 1 < 10 && step1 >= locol && step1 <= hicol) ? 1.0f : 0.0f;
                const v2f av = { afrag[ch][q][0], afrag[ch][q][1] };
                const v2f bv = { b0, b1 };
                acc = __builtin_amdgcn_wmma_f32_16x16x4_f32(
                    /*neg_a=*/false, av, /*neg_b=*/false, bv,
                    /*c_mod=*/(short)0, acc, /*reuse_a=*/false, /*reuse_b=*/false);
            }
            // D[M][N]: VGPR r, lane l -> M = r + 8*(l>=16), N = l&15
#pragma unroll
            for (int r = 0; r < 8; ++r) dump[wave][r * 32 + lane] = acc[r];
            __syncthreads();
            const float sval = dump[wave][(bcol & 7) * 32 + ((bcol >> 3) << 4) + c]; // row b
            const float cval = dump[wave][2 * 32 + 16 + c];                          // row 10 (count)
            __syncthreads();
            if ((lane >> 4) == tile) ratio[ch] = sval / cval;
        }
    }

    out[n] = ratio[0] * ratio[1];
}

extern "C" void kernel_launch(void* const* d_in, const int* in_sizes, int n_in,
                              void* d_out, int out_size, void* d_ws, size_t ws_size,
                              hipStream_t stream) {
    // setup_inputs() order:
    // 0 inptasksobs(unused) 1 inptasksperf 2 inptaskspred(unused) 3 num_obs_tasks(unused)
    // 4 tasksobsids(unused) 5 taskspredids(unused) 6 obs_sens 7 pred_sens 8 obs_proc
    // 9 pred_proc 10 betas 11 zetas
    const int*   perf      = (const int*)d_in[1];
    const float* obs_sens  = (const float*)d_in[6];
    const float* pred_sens = (const float*)d_in[7];
    const float* obs_proc  = (const float*)d_in[8];
    const float* pred_proc = (const float*)d_in[9];
    const float* betas     = (const float*)d_in[10];
    const float* zetas     = (const float*)d_in[11];
    float* outp = (float*)d_out;

    dim3 grid(NTOT_ / BLK_), block(BLK_);
    hipLaunchKernelGGL(trust_kernel, grid, block, 0, stream,
                       perf, obs_sens, obs_proc, pred_sens, pred_proc, betas, zetas, outp);
}